// YoloLoss_57088705298620
// MI455X (gfx1250) — compile-verified
//
#include <hip/hip_runtime.h>
#include <hip/hip_bf16.h>
#include <math.h>

typedef float v2f __attribute__((ext_vector_type(2)));
typedef float v8f __attribute__((ext_vector_type(8)));

#define NBATCH 16
#define NANCH  5
#define NH     128
#define NW     128
#define NCLS   20
#define CH     26
#define MAXT   50
#define NCELLS (NBATCH*NANCH*NH*NW)   // 1,310,720

// ---------- helpers ----------
__device__ __forceinline__ float softplusf(float x) {
  // jax.nn.softplus, numerically stable
  return fmaxf(x, 0.0f) + log1pf(expf(-fabsf(x)));
}
__device__ __forceinline__ float inv_tanhf(float y) {
  float yc = fminf(fmaxf(y, -1.0f + 1e-6f), 1.0f - 1e-6f);
  float v = 0.5f * logf((1.0f + yc) / (1.0f - yc));
  return (y <= -1.0f) ? -2.0f : ((y >= 1.0f) ? 2.0f : v);
}

// acc slots:
// 0 sx, 1 sy, 2 sw, 3 sh, 4 bce_true, 5 ce, 6 nM, 7 bceF_minus,
// 8 nF_minus, 9 nGT, 10 nCorrect, 11 softplus_total, 12 nProposals
__global__ void yolo_zero(float* acc) {
  if (threadIdx.x < 32) acc[threadIdx.x] = 0.0f;
}

// ---------- Phase A: exact sequential target scan, one block per batch ----------
__global__ void yolo_phaseA(const float* __restrict__ pred,
                            const float* __restrict__ target,
                            const int*   __restrict__ tsizes,
                            const float* __restrict__ anchors,
                            float* __restrict__ acc) {
  __shared__ int   e_cell[260];
  __shared__ int   e_cm[260];
  __shared__ int   e_mk[260];
  __shared__ int   e_lb[260];
  __shared__ float e_tx[260], e_ty[260], e_tw[260], e_th[260];
  if (threadIdx.x != 0) return;
  const int b = blockIdx.x;

  float aw[NANCH], ah[NANCH];
  for (int a = 0; a < NANCH; ++a) { aw[a] = anchors[2*a] * 0.125f; ah[a] = anchors[2*a+1] * 0.125f; }

  int nent = 0;
  auto findOrCreate = [&](int cell) -> int {
    for (int k = 0; k < nent; ++k) if (e_cell[k] == cell) return k;
    int k = nent++;
    e_cell[k] = cell; e_cm[k] = 1; e_mk[k] = 0; e_lb[k] = 0;
    e_tx[k] = 0.0f; e_ty[k] = 0.0f; e_tw[k] = 0.0f; e_th[k] = 0.0f;
    return k;
  };

  float nGT = 0.0f, nCor = 0.0f;
  const int ts = tsizes[b];

  for (int t = 0; t < MAXT; ++t) {
    const float* row = target + ((size_t)(b*MAXT + t)) * 33;
    float gx = row[0]*0.125f, gy = row[1]*0.125f;
    float gh = row[3]*0.125f, gw = row[4]*0.125f;
    bool valid = (t < ts) && (gw != 0.0f) && (gh != 0.0f);
    if (!valid) continue;  // all updates are where(valid,...) no-ops

    int gi = (int)gx; gi = gi < 0 ? 0 : (gi > NW-1 ? NW-1 : gi);
    int gj = (int)gy; gj = gj < 0 ? 0 : (gj > NH-1 ? NH-1 : gj);

    float best = -1e30f; int bn = 0; float ious[NANCH];
    for (int a = 0; a < NANCH; ++a) {
      float inter = fmaxf(fminf(gw, aw[a]) + 1.0f, 0.0f) *
                    fmaxf(fminf(gh, ah[a]) + 1.0f, 0.0f);
      float iou = inter / ((gw+1.0f)*(gh+1.0f) + (aw[a]+1.0f)*(ah[a]+1.0f) - inter + 1e-16f);
      ious[a] = iou;
      if (iou > best) { best = iou; bn = a; }   // strict > == first-max like jnp.argmax
    }
    // event 1: conf_mask := 0 where anch_iou > IGNORE_THRES
    for (int a = 0; a < NANCH; ++a)
      if (ious[a] > 0.5f) { int k = findOrCreate((a*NH + gj)*NW + gi); e_cm[k] = 0; }
    // event 2: set masked-cell values at (bn,gj,gi)  (after event 1 in same step)
    int k = findOrCreate((bn*NH + gj)*NW + gi);
    e_cm[k] = 1; e_mk[k] = 1;
    e_tx[k] = inv_tanhf(gx - ((float)gi + 0.5f));
    e_ty[k] = inv_tanhf(gy - ((float)gj + 0.5f));
    e_tw[k] = logf(gw / aw[bn] + 1e-16f);
    e_th[k] = logf(gh / ah[bn] + 1e-16f);
    int lb = 0; float bm = row[13];
    for (int c = 1; c < NCLS; ++c) { float v = row[13+c]; if (v > bm) { bm = v; lb = c; } }
    e_lb[k] = lb;
    nGT += 1.0f;

    // nCorrect: iou(gbox, pred_box) > .5 && class match && conf > 0
    size_t base = ((((size_t)b*NANCH + bn)*NH + gj)*NW + gi) * CH;
    float conf = pred[base+0];
    float px  = tanhf(pred[base+1]) + 0.5f + (float)gi;
    float py  = tanhf(pred[base+2]) + 0.5f + (float)gj;
    float phh = expf(pred[base+4]) * ah[bn];
    float pww = expf(pred[base+5]) * aw[bn];
    float b1x1 = gx-gw,  b1x2 = gx+gw,  b1y1 = gy-gh,  b1y2 = gy+gh;
    float b2x1 = px-pww, b2x2 = px+pww, b2y1 = py-phh, b2y2 = py+phh;
    float iw = fmaxf(fminf(b1x2,b2x2) - fmaxf(b1x1,b2x1) + 1.0f, 0.0f);
    float ih = fmaxf(fminf(b1y2,b2y2) - fmaxf(b1y1,b2y1) + 1.0f, 0.0f);
    float inter = iw*ih;
    float a1 = (b1x2-b1x1+1.0f)*(b1y2-b1y1+1.0f);
    float a2 = (b2x2-b2x1+1.0f)*(b2y2-b2y1+1.0f);
    float iou = inter/(a1+a2-inter+1e-16f);
    int pc = 0; float pbm = pred[base+6];
    for (int c = 1; c < NCLS; ++c) { float v = pred[base+6+c]; if (v > pbm) { pbm = v; pc = c; } }
    if (iou > 0.5f && pc == lb && conf > 0.0f) nCor += 1.0f;
  }

  // finalize entries (every touched cell ends with cm_false == 0)
  float sx=0,sy=0,sw=0,sh=0,bceT=0,ceS=0,nM=0,bceFm=0,nFm=0;
  for (int k = 0; k < nent; ++k) {
    size_t base = ((size_t)b*NANCH*NH*NW + (size_t)e_cell[k]) * CH;
    float conf = pred[base+0];
    bceFm += softplusf(conf);   // what phase B over-counts at this cell
    nFm   += 1.0f;
    if (e_mk[k]) {
      nM += 1.0f;
      float dx = pred[base+1]-e_tx[k]; sx += dx*dx;
      float dy = pred[base+2]-e_ty[k]; sy += dy*dy;
      float dh = pred[base+4]-e_th[k]; sh += dh*dh;
      float dw = pred[base+5]-e_tw[k]; sw += dw*dw;
      bceT += softplusf(conf) - conf;     // tconf = 1 here
      float m = pred[base+6];
      for (int c = 1; c < NCLS; ++c) m = fmaxf(m, pred[base+6+c]);
      float s = 0.0f;
      for (int c = 0; c < NCLS; ++c) s += expf(pred[base+6+c] - m);
      ceS += (m + logf(s)) - pred[base+6+e_lb[k]];
    }
  }
  atomicAdd(&acc[0], sx);   atomicAdd(&acc[1], sy);
  atomicAdd(&acc[2], sw);   atomicAdd(&acc[3], sh);
  atomicAdd(&acc[4], bceT); atomicAdd(&acc[5], ceS);
  atomicAdd(&acc[6], nM);   atomicAdd(&acc[7], bceFm);
  atomicAdd(&acc[8], nFm);  atomicAdd(&acc[9], nGT);
  atomicAdd(&acc[10], nCor);
}

// ---------- Phase B: dense conf-channel streaming + WMMA block reduction ----------
__global__ __launch_bounds__(256)
void yolo_phaseB(const float* __restrict__ pred, float* __restrict__ acc) {
  const int stride = (int)(gridDim.x * blockDim.x);
  float sp = 0.0f, np = 0.0f;
  for (int i = (int)(blockIdx.x * blockDim.x + threadIdx.x); i < NCELLS; i += stride) {
    __builtin_prefetch(pred + (size_t)(i + stride) * CH, 0, 0);  // global_prefetch_b8
    float conf = pred[(size_t)i * CH];
    sp += softplusf(conf);
    np += (conf > 0.0f) ? 1.0f : 0.0f;
  }

  // 256 threads x 16 accumulator slots -> reduce with V_WMMA_F32_16X16X4_F32
  __shared__ float red[256 * 16];
  const int tid = threadIdx.x;
  #pragma unroll
  for (int k = 0; k < 16; ++k) red[tid*16 + k] = 0.0f;
  red[tid*16 + 11] = sp;
  red[tid*16 + 12] = np;
  __syncthreads();

  const int lane = tid & 31;
  const int wrow = (tid >> 5) * 32;     // each wave reduces its own 32 rows
  const int n    = lane & 15;           // column (accumulator slot)
  const int hi   = (lane >> 4) * 2;     // lanes 16-31 take K=2,3 per A-layout symmetry
  v8f c = {};
  v2f aones; aones[0] = 1.0f; aones[1] = 1.0f;   // A = ones(16x4)  =>  D[m][n] = sum_k B[k][n]
  #pragma unroll
  for (int ch = 0; ch < 8; ++ch) {
    int r = wrow + ch*4 + hi;
    v2f bv; bv[0] = red[(r+0)*16 + n]; bv[1] = red[(r+1)*16 + n];
    c = __builtin_amdgcn_wmma_f32_16x16x4_f32(false, aones, false, bv, (short)0, c,
                                              false, false);
  }
  // all D rows are identical column sums; lanes 0..15, VGPR0 hold D[*][n]
  if (lane < 16) atomicAdd(&acc[lane], c[0]);
}

// ---------- Phase C: final scalar combine ----------
__global__ void yolo_phaseC(const float* __restrict__ acc, float* __restrict__ out) {
  if (threadIdx.x != 0 || blockIdx.x != 0) return;
  const float Ntot = (float)NCELLS;
  float nM = fmaxf(acc[6], 1.0f);
  float nF = fmaxf(Ntot - acc[8], 1.0f);
  float loss_x = acc[0]/nM, loss_y = acc[1]/nM, loss_w = acc[2]/nM, loss_h = acc[3]/nM;
  float bceF = acc[11] - acc[7];               // dense softplus sum minus touched cells
  float loss_conf = 1.25f * bceF / nF + acc[4] / nM;
  float loss_cls  = (1.0f / (float)NBATCH) * acc[5] / nM;
  float loss = loss_x + loss_y + loss_w + loss_h + loss_conf + loss_cls;
  float nGT = acc[9], nCor = acc[10], nProp = acc[12];
  out[0] = loss;
  out[1] = (nGT  > 0.0f) ? nCor / fmaxf(nGT,  1.0f) : 1.0f;
  out[2] = (nProp> 0.0f) ? nCor / fmaxf(nProp,1.0f) : 1.0f;
}

extern "C" void kernel_launch(void* const* d_in, const int* in_sizes, int n_in,
                              void* d_out, int out_size, void* d_ws, size_t ws_size,
                              hipStream_t stream) {
  const float* pred    = (const float*)d_in[0];   // (16,5,128,128,26) f32
  const float* target  = (const float*)d_in[1];   // (16,50,33) f32
  const int*   tsizes  = (const int*)  d_in[2];   // (16,) i32
  const float* anchors = (const float*)d_in[3];   // (5,2) f32
  float* out = (float*)d_out;
  float* acc = (float*)d_ws;                      // 32 floats of scratch

  yolo_zero  <<<1, 32, 0, stream>>>(acc);
  yolo_phaseA<<<NBATCH, 32, 0, stream>>>(pred, target, tsizes, anchors, acc);
  yolo_phaseB<<<512, 256, 0, stream>>>(pred, acc);
  yolo_phaseC<<<1, 32, 0, stream>>>(acc, out);
}